// Bi_Self_RNN_31301721653611
// MI455X (gfx1250) — compile-verified
//
#include <hip/hip_runtime.h>
#include <math.h>

// ---------------------------------------------------------------------------
// Bi_Self_RNN for MI455X (gfx1250, wave32).
//
// 3 waves per workgroup own 16 batch rows and run the 199-step recurrence with
// carries in LDS. Wave w holds the B-fragments of exactly one weight matrix
// (w0:Wq, w1:Wk, w2:Wv) in 128 VGPRs, so the per-step critical path is only
// 5 projections (was 14 with a single wave). All 16x64 @ 64x64 projections use
// V_WMMA_F32_16X16X4_F32 (exact f32). d_ws holds the pre-swizzled weights.
// ---------------------------------------------------------------------------

#define LS     68            // LDS row stride (floats); conflict-free banks
#define BUF    (16 * LS)     // one 16x64 f32 buffer (padded)
#define TSTEPS 200
#define DDIM   64

typedef float v2f __attribute__((ext_vector_type(2)));
typedef float v8f __attribute__((ext_vector_type(8)));

__device__ __forceinline__ float fast_tanh(float v) {
#if __has_builtin(__builtin_amdgcn_tanhf)
  return __builtin_amdgcn_tanhf(v);
#elif __has_builtin(__builtin_amdgcn_tanh_f32)
  return __builtin_amdgcn_tanh_f32(v);
#else
  return tanhf(v);
#endif
}

__device__ __forceinline__ v8f wmma_f32(v2f a, v2f b, v8f c) {
  // (neg_a, A, neg_b, B, c_mod, C, reuse_a, reuse_b)
  return __builtin_amdgcn_wmma_f32_16x16x4_f32(false, a, false, b, (short)0, c,
                                               false, false);
}

// dst(16x64) = src(16x64) @ W(64x64), exact f32, W held as 64 register-
// resident B-fragments (wf[kc*4+nt], one v2f per lane).
// A fragment (f32 16x4): lane L holds (M=L&15, K=4*kc + j + 2*(L>>4)) -> a
// contiguous ds_load_b64 from the row-major LDS buffer.
// C/D (f32 16x16): VGPR r, lanes 0-15 -> M=r, lanes 16-31 -> M=r+8, N=lane&15.
__device__ __forceinline__ void proj16x64(const float* src, const v2f* wf,
                                          float* dst, int m, int h) {
  v8f a0 = {}, a1 = {}, a2 = {}, a3 = {};
  const float* arow = src + m * LS + 2 * h;
#pragma unroll
  for (int kc = 0; kc < 16; ++kc) {
    v2f af = *(const v2f*)(arow + 4 * kc);          // ds_load_b64
    a0 = wmma_f32(af, wf[kc * 4 + 0], a0);          // 4 independent acc chains
    a1 = wmma_f32(af, wf[kc * 4 + 1], a1);
    a2 = wmma_f32(af, wf[kc * 4 + 2], a2);
    a3 = wmma_f32(af, wf[kc * 4 + 3], a3);
  }
#pragma unroll
  for (int r = 0; r < 8; ++r) {
    int row = r + 8 * h;
    dst[row * LS + m +  0] = a0[r];
    dst[row * LS + m + 16] = a1[r];
    dst[row * LS + m + 32] = a2[r];
    dst[row * LS + m + 48] = a3[r];
  }
}

// Row-wise 64-length dot: lane (m,h) sums its 32-col half, then adds the
// partner half via a cross-lane xor-16 shuffle (both lanes in same wave32).
__device__ __forceinline__ float rowdot(const float* A, const float* B, int m,
                                        int h) {
  const float4* a = (const float4*)(A + m * LS + 32 * h);
  const float4* b = (const float4*)(B + m * LS + 32 * h);
  float s = 0.f;
#pragma unroll
  for (int i = 0; i < 8; ++i) {
    float4 u = a[i], v = b[i];
    s += u.x * v.x + u.y * v.y + u.z * v.z + u.w * v.w;
  }
  s += __shfl_xor(s, 16, 32);
  return s;
}

__device__ __forceinline__ void softmax2(float a, float b, float& p0,
                                         float& p1) {
  float mx  = fmaxf(a, b);
  float e0  = __expf(a - mx);
  float e1  = __expf(b - mx);
  float inv = 1.f / (e0 + e1);
  p0 = e0 * inv;
  p1 = e1 * inv;
}

// dst row = p0*V0 row + p1*V1 row (optionally tanh'ed), row-per-lane-half.
template <bool DO_TANH>
__device__ __forceinline__ void combine(const float* V0, const float* V1,
                                        float p0, float p1, float* dst, int m,
                                        int h) {
  const float4* v0 = (const float4*)(V0 + m * LS + 32 * h);
  const float4* v1 = (const float4*)(V1 + m * LS + 32 * h);
  float4* d = (float4*)(dst + m * LS + 32 * h);
#pragma unroll
  for (int i = 0; i < 8; ++i) {
    float4 a = v0[i], b = v1[i], r;
    r.x = p0 * a.x + p1 * b.x;
    r.y = p0 * a.y + p1 * b.y;
    r.z = p0 * a.z + p1 * b.z;
    r.w = p0 * a.w + p1 * b.w;
    if (DO_TANH) {
      r.x = fast_tanh(r.x);
      r.y = fast_tanh(r.y);
      r.z = fast_tanh(r.z);
      r.w = fast_tanh(r.w);
    }
    d[i] = r;
  }
}

// Pre-pass: swizzle Wq/Wk/Wv (64x64 row-major, (d,u)) into WMMA B-fragment
// order: idx = (((w*16 + kc)*4 + nt)*32 + lane)*2 + j  ->  W[k][n] with
// k = 4*kc + j + 2*(lane>>4), n = 16*nt + (lane&15). 48 KB total, L2-resident.
__global__ void swizzle_weights(const float* __restrict__ Wq,
                                const float* __restrict__ Wk,
                                const float* __restrict__ Wv,
                                float* __restrict__ wsw) {
  int idx = blockIdx.x * blockDim.x + threadIdx.x;
  if (idx >= 3 * 16 * 4 * 32 * 2) return;
  int j    = idx & 1;
  int lane = (idx >> 1) & 31;
  int nt   = (idx >> 6) & 3;
  int kc   = (idx >> 8) & 15;
  int w    = idx >> 12;
  int k = 4 * kc + j + 2 * (lane >> 4);
  int n = 16 * nt + (lane & 15);
  const float* W = (w == 0) ? Wq : (w == 1) ? Wk : Wv;
  wsw[idx] = W[k * 64 + n];
}

__global__ __launch_bounds__(96) void birnn_kernel(
    const float* __restrict__ x, const float* __restrict__ wsw,
    float* __restrict__ out) {
  __shared__ __align__(16) float smem[13 * BUF];
  float* bufL  = smem + 0 * BUF;   // carry l  (and final output a0)
  float* bufS  = smem + 1 * BUF;   // carry s
  float* bufX  = smem + 2 * BUF;   // x_t
  float* bufO0 = smem + 3 * BUF;   // o0 = tanh(attn1 row 0)
  float* bufO1 = smem + 4 * BUF;   // o1 = tanh(attn2 row 1)
  float* P0 = smem + 5 * BUF;      // Ql | q0
  float* P1 = smem + 6 * BUF;      // Kl | k0
  float* P2 = smem + 7 * BUF;      // Vl | v0
  float* P3 = smem + 8 * BUF;      // Qx | q1
  float* P4 = smem + 9 * BUF;      // Kx | k1
  float* P5 = smem + 10 * BUF;     // Vx | v1
  float* P6 = smem + 11 * BUF;     // Ks
  float* P7 = smem + 12 * BUF;     // Vs

  const int lane = threadIdx.x & 31;  // lane within wave
  const int wid  = threadIdx.x >> 5;  // wave id: 0 -> Wq, 1 -> Wk, 2 -> Wv
  const int m = lane & 15;            // batch row within the group of 16
  const int h = lane >> 4;            // column half (0: 0-31, 1: 32-63)
  const int b = blockIdx.x * 16 + m;
  const float kScale = 0.125f;        // 1/sqrt(64)

  // This wave's weight matrix as 64 register-resident B-fragments (128 VGPRs).
  v2f wf[64];
  {
    const float* myw = wsw + wid * 4096 + lane * 2;
#pragma unroll
    for (int f = 0; f < 64; ++f) wf[f] = *(const v2f*)(myw + f * 64);
  }

  // carries: l = s = x[:, 0]
  if (wid == 0) {
    const float4* xr = (const float4*)(x + (size_t)b * TSTEPS * DDIM + 32 * h);
    float4* dl  = (float4*)(bufL + m * LS + 32 * h);
    float4* dsv = (float4*)(bufS + m * LS + 32 * h);
#pragma unroll
    for (int i = 0; i < 8; ++i) {
      float4 v = xr[i];
      dl[i]  = v;
      dsv[i] = v;
    }
  }
  __syncthreads();

#pragma unroll 1
  for (int t = 1; t < TSTEPS; ++t) {
    // stage x_t into LDS (wave 2); prefetch next timestep's row
    if (wid == 2) {
      const float* xrow = x + ((size_t)b * TSTEPS + t) * DDIM + 32 * h;
      const float4* xr = (const float4*)xrow;
      float4* dx = (float4*)(bufX + m * LS + 32 * h);
#pragma unroll
      for (int i = 0; i < 8; ++i) dx[i] = xr[i];
      if (t + 1 < TSTEPS) __builtin_prefetch(xrow + DDIM, 0, 0);
    }
    __syncthreads();

    // ---- phase 1: projections, split by weight matrix across waves ----
    if (wid == 0) {                       // Q projections
      proj16x64(bufL, wf, P0, m, h);      // Ql = l  @ Wq
      proj16x64(bufX, wf, P3, m, h);      // Qx = xt @ Wq
    } else if (wid == 1) {                // K projections
      proj16x64(bufL, wf, P1, m, h);      // Kl = l  @ Wk
      proj16x64(bufX, wf, P4, m, h);      // Kx = xt @ Wk
      proj16x64(bufS, wf, P6, m, h);      // Ks = s  @ Wk
    } else {                              // V projections
      proj16x64(bufL, wf, P2, m, h);      // Vl = l  @ Wv
      proj16x64(bufX, wf, P5, m, h);      // Vx = xt @ Wv
      proj16x64(bufS, wf, P7, m, h);      // Vs = s  @ Wv
    }
    __syncthreads();

    // attn1 row0 (wave0): softmax([ql.kl, ql.kx]/8) @ [Vl; Vx], tanh -> o0
    // attn2 row1 (wave1): softmax([qx.ks, qx.kx]/8) @ [Vs; Vx], tanh -> o1
    if (wid == 0) {
      float d00 = rowdot(P0, P1, m, h) * kScale;
      float d01 = rowdot(P0, P4, m, h) * kScale;
      float p0, p1;
      softmax2(d00, d01, p0, p1);
      combine<true>(P2, P5, p0, p1, bufO0, m, h);
    } else if (wid == 1) {
      float d10 = rowdot(P3, P6, m, h) * kScale;
      float d11 = rowdot(P3, P4, m, h) * kScale;
      float q0, q1;
      softmax2(d10, d11, q0, q1);
      combine<true>(P7, P5, q0, q1, bufO1, m, h);
    }
    __syncthreads();

    // ---- phase 2: attn3 over o = [o0; o1], split by weight matrix ----
    if (wid == 0) {
      proj16x64(bufO0, wf, P0, m, h);     // q0
      proj16x64(bufO1, wf, P3, m, h);     // q1
    } else if (wid == 1) {
      proj16x64(bufO0, wf, P1, m, h);     // k0
      proj16x64(bufO1, wf, P4, m, h);     // k1
    } else {
      proj16x64(bufO0, wf, P2, m, h);     // v0
      proj16x64(bufO1, wf, P5, m, h);     // v1
    }
    __syncthreads();

    if (wid == 0) {
      float e00 = rowdot(P0, P1, m, h) * kScale;
      float e01 = rowdot(P0, P4, m, h) * kScale;
      float r0, r1;
      softmax2(e00, e01, r0, r1);
      combine<false>(P2, P5, r0, r1, bufL, m, h);   // a0 -> new l (also y_t)
    } else if (wid == 1) {
      float e10 = rowdot(P3, P1, m, h) * kScale;
      float e11 = rowdot(P3, P4, m, h) * kScale;
      float s0, s1;
      softmax2(e10, e11, s0, s1);
      combine<false>(P2, P5, s0, s1, bufS, m, h);   // a1 -> new s
    }
    __syncthreads();
  }

  // output = ys[-1] = final a0, held in bufL
  if (wid == 0) {
    const float4* src = (const float4*)(bufL + m * LS + 32 * h);
    float4* dst = (float4*)(out + (size_t)b * DDIM + 32 * h);
#pragma unroll
    for (int i = 0; i < 8; ++i) dst[i] = src[i];
  }
}

extern "C" void kernel_launch(void* const* d_in, const int* in_sizes, int n_in,
                              void* d_out, int out_size, void* d_ws,
                              size_t ws_size, hipStream_t stream) {
  (void)in_sizes; (void)n_in; (void)out_size; (void)ws_size;
  const float* x  = (const float*)d_in[0];
  const float* Wq = (const float*)d_in[1];
  const float* Wk = (const float*)d_in[2];
  const float* Wv = (const float*)d_in[3];
  float* wsw = (float*)d_ws;  // 3 * 4096 floats = 48 KB swizzled weights

  swizzle_weights<<<48, 256, 0, stream>>>(Wq, Wk, Wv, wsw);
  birnn_kernel<<<2048 / 16, 96, 0, stream>>>(x, wsw, (float*)d_out);
}